// PatchFilter_82291573391646
// MI455X (gfx1250) — compile-verified
//
#include <hip/hip_runtime.h>
#include <hip/hip_bf16.h>
#include <math.h>

// ---------------------------------------------------------------------------
// PatchFilter on MI455X (gfx1250, wave32)
//
// Shapes: image (8, 32, 257, 1024) f32 -> out (8, 288, 1024) f32
// Total traffic ~43MB => ~1.9us at 23.3TB/s; compute ~3.4 GFLOP => the
// workload is latency-bound. We use fp32 WMMA (V_WMMA_F32_16X16X4_F32) for
// the 768x64xK=1024 similarity GEMM because the sims feed argmax/top-k
// (discrete decisions): bf16/fp8 quantization would flip borderline
// assignments and corrupt the pooled means, while f32 WMMA matches the
// fp32 reference.
// ---------------------------------------------------------------------------

#define B_    8
#define T_    32
#define P1_   257
#define P_    256
#define D_    1024
#define NG    4
#define SG    64
#define NTOK  768          // WIN * P_
#define OUTT  (T_ + NG*SG) // 288

// selected = round(linspace(0,31,4)) = {0,10,21,31}; starts clamp last to 29
__constant__ int c_sel[NG]   = {0, 10, 21, 31};
__constant__ int c_start[NG] = {0, 10, 21, 29};

typedef __attribute__((ext_vector_type(2))) float v2f;
typedef __attribute__((ext_vector_type(8))) float v8f;

__device__ __forceinline__ const float* token_row(const float* img, int b, int start, int t) {
    int f = t >> 8;          // frame within window (0..2)
    int p = t & (P_ - 1);    // patch (0..255)
    return img + (size_t)((b * T_ + start + f) * P1_ + 1 + p) * D_;
}

__device__ __forceinline__ float wave_sum(float v) {
    #pragma unroll
    for (int m = 16; m >= 1; m >>= 1) v += __shfl_xor(v, m, 32);
    return v;
}

// ---------------------------------------------------------------------------
// Kernel 1: copy cls tokens to out[:, 0:32, :]
// ---------------------------------------------------------------------------
__global__ void copy_cls_kernel(const float* __restrict__ img, float* __restrict__ out) {
    int idx = blockIdx.x * blockDim.x + threadIdx.x;   // over B*T*(D/4)
    int total = B_ * T_ * (D_ / 4);
    if (idx >= total) return;
    int d4 = idx % (D_ / 4);
    int bt = idx / (D_ / 4);
    int t = bt % T_, b = bt / T_;
    const float4* src = (const float4*)(img + (size_t)((b * T_ + t) * P1_) * D_) + d4;
    float4*       dst = (float4*)(out + (size_t)(b * OUTT + t) * D_) + d4;
    *dst = *src;
}

// ---------------------------------------------------------------------------
// Kernel 2: per group (b,g): token norms, keep flags, cls sims, exact top-64
// One block (256 threads = 8 waves) per group; wave-per-row reductions.
// ---------------------------------------------------------------------------
__global__ void prep_kernel(const float* __restrict__ img,
                            float* __restrict__ ws_norm,
                            int*   __restrict__ ws_valid,
                            int*   __restrict__ ws_topk) {
    int gid = blockIdx.x;            // 0..31  == b*NG + g
    int b = gid >> 2, g = gid & 3;
    int start = c_start[g], sel = c_sel[g];
    float* norms = ws_norm + gid * NTOK;
    int*   valid = ws_valid + gid * NTOK;
    int*   topk  = ws_topk  + gid * SG;

    __shared__ float s_norm[NTOK + 1];   // token norms + cls norm
    __shared__ float s_sim[P_];

    int tid = threadIdx.x;
    int wave = tid >> 5, lane = tid & 31;
    const float* clsrow = img + (size_t)((b * T_ + sel) * P1_) * D_;  // patch 0 of selected frame

    // ---- norms of 768 tokens + cls vector ----
    for (int r = wave; r < NTOK + 1; r += 8) {
        const float* row = (r < NTOK) ? token_row(img, b, start, r) : clsrow;
        float s = 0.f;
        for (int j = lane; j < D_; j += 32) { float x = row[j]; s = fmaf(x, x, s); }
        s = wave_sum(s);
        if (lane == 0) {
            float n = sqrtf(s);
            s_norm[r] = n;
            if (r < NTOK) norms[r] = n;
        }
    }
    __syncthreads();

    // ---- keep flags: cossim(IDR[p], win[f][p]) < 0.7 for f=1,2 ----
    for (int i = wave; i < 2 * P_; i += 8) {
        int t = P_ + i;
        int p = i & (P_ - 1);
        const float* a = token_row(img, b, start, p);
        const float* c = token_row(img, b, start, t);
        float s = 0.f;
        for (int j = lane; j < D_; j += 32) s = fmaf(a[j], c[j], s);
        s = wave_sum(s);
        if (lane == 0) {
            float den = fmaxf(s_norm[p], 1e-8f) * fmaxf(s_norm[t], 1e-8f);
            valid[t] = ((s / den) < 0.7f) ? 1 : 0;
        }
    }
    if (tid < P_) valid[tid] = 1;   // all IDR tokens valid

    // ---- cls_sim[p] = cossim(IDR[p], cls) ----
    for (int p = wave; p < P_; p += 8) {
        const float* a = token_row(img, b, start, p);
        float s = 0.f;
        for (int j = lane; j < D_; j += 32) s = fmaf(a[j], clsrow[j], s);
        s = wave_sum(s);
        if (lane == 0) {
            float den = fmaxf(s_norm[p], 1e-8f) * fmaxf(s_norm[NTOK], 1e-8f);
            s_sim[p] = s / den;
        }
    }
    __syncthreads();

    // ---- exact jax.lax.top_k(·, 64): rank = #(strictly greater) + #(equal, lower idx)
    if (tid < P_) {
        float v = s_sim[tid];
        int rank = 0;
        for (int q = 0; q < P_; ++q) {
            float u = s_sim[q];
            if (u > v || (u == v && q < tid)) rank++;
        }
        if (rank < SG) topk[rank] = tid;
    }
}

// ---------------------------------------------------------------------------
// Kernel 3: similarity GEMM (768x64, K=1024) via V_WMMA_F32_16X16X4_F32 +
// per-token argmax over centers (lowest-index tie-break, like jnp.argmax).
// One wave per 16-token row tile; A loaded once per k-step, reused across
// the 4 center col-tiles (4 live v8f accumulators).
//
// A layout (16x4 f32): lanes 0-15 row M=lane K={k,k+1}; lanes 16-31 K={k+2,k+3}
// B layout (4x16 f32): mirrored with N=lane&15
// C layout (16x16 f32): VGPR r -> M = r + 8*(lane>=16), N = lane&15
// ---------------------------------------------------------------------------
__global__ void simgemm_kernel(const float* __restrict__ img,
                               const float* __restrict__ ws_norm,
                               const int*   __restrict__ ws_topk,
                               int*         __restrict__ ws_assigned) {
    int tile = blockIdx.x;           // 0..47 (16-token row tiles; never straddle frames)
    int g = blockIdx.y, b = blockIdx.z;
    int gid = b * NG + g;
    int start = c_start[g];
    const float* norms = ws_norm + gid * NTOK;
    const int*   topk  = ws_topk + gid * SG;
    int*      assigned = ws_assigned + gid * NTOK;

    int lane = threadIdx.x;          // 0..31 (full wave, EXEC all-ones for WMMA)
    int half = lane >> 4;
    int l16  = lane & 15;

    int rowBase = tile * 16;
    const float* arow = token_row(img, b, start, rowBase + l16) + 2 * half;
    const float* idr0 = token_row(img, b, start, 0);

    // reciprocal norms for this lane's C rows (M = r + 8*half)
    float rnT[8];
    #pragma unroll
    for (int r = 0; r < 8; ++r)
        rnT[r] = 1.0f / fmaxf(norms[rowBase + r + 8 * half], 1e-12f);

    // center pointers / norms for the 4 col-tiles (N = ct*16 + l16)
    const float* brow[4];
    float rnC[4];
    v8f acc[4];
    #pragma unroll
    for (int ct = 0; ct < 4; ++ct) {
        int cidx = topk[ct * 16 + l16];
        brow[ct] = idr0 + (size_t)cidx * D_ + 2 * half;
        rnC[ct]  = 1.0f / fmaxf(norms[cidx], 1e-12f);
        v8f z = {0.f, 0.f, 0.f, 0.f, 0.f, 0.f, 0.f, 0.f};
        acc[ct] = z;
    }

    // K loop: one A load per step, 4 WMMAs (independent accumulators)
    #pragma unroll 2
    for (int k = 0; k < D_; k += 4) {
        v2f a = *(const v2f*)(arow + k);
        #pragma unroll
        for (int ct = 0; ct < 4; ++ct) {
            v2f bb = *(const v2f*)(brow[ct] + k);
            acc[ct] = __builtin_amdgcn_wmma_f32_16x16x4_f32(
                false, a, false, bb, (short)0, acc[ct], false, false);
        }
    }

    // per-lane argmax across col-tiles (n grows with ct -> equal keeps lower n)
    float bestv[8]; int bestn[8];
    #pragma unroll
    for (int r = 0; r < 8; ++r) { bestv[r] = -INFINITY; bestn[r] = SG; }
    #pragma unroll
    for (int ct = 0; ct < 4; ++ct) {
        int n = ct * 16 + l16;
        #pragma unroll
        for (int r = 0; r < 8; ++r) {
            float val = acc[ct][r] * rnT[r] * rnC[ct];
            if (val > bestv[r] || (val == bestv[r] && n < bestn[r])) {
                bestv[r] = val; bestn[r] = n;
            }
        }
    }

    // cross-lane argmax within each 16-lane half (masks < 16 stay in-half)
    #pragma unroll
    for (int r = 0; r < 8; ++r) {
        float bv = bestv[r]; int bn = bestn[r];
        #pragma unroll
        for (int m = 1; m < 16; m <<= 1) {
            float ov = __shfl_xor(bv, m, 32);
            int   on = __shfl_xor(bn, m, 32);
            if (ov > bv || (ov == bv && on < bn)) { bv = ov; bn = on; }
        }
        bestv[r] = bv; bestn[r] = bn;
    }
    if (l16 == 0) {
        #pragma unroll
        for (int r = 0; r < 8; ++r)
            assigned[rowBase + r + 8 * half] = bestn[r];
    }
}

// ---------------------------------------------------------------------------
// Kernel 4: masked mean-pool per (center, group); deterministic, no atomics.
// Block: 256 threads, each owns 4 consecutive output columns (float4).
// ---------------------------------------------------------------------------
__global__ void pool_kernel(const float* __restrict__ img,
                            const int* __restrict__ ws_valid,
                            const int* __restrict__ ws_topk,
                            const int* __restrict__ ws_assigned,
                            float* __restrict__ out) {
    int c = blockIdx.x;              // 0..63
    int g = blockIdx.y, b = blockIdx.z;
    int gid = b * NG + g;
    int start = c_start[g];
    const int* valid    = ws_valid    + gid * NTOK;
    const int* assigned = ws_assigned + gid * NTOK;
    const int* topk     = ws_topk     + gid * SG;

    __shared__ int s_match[NTOK];
    int tid = threadIdx.x;
    for (int t = tid; t < NTOK; t += 256)
        s_match[t] = (assigned[t] == c) && valid[t];
    __syncthreads();

    int col = tid * 4;
    float4 acc = {0.f, 0.f, 0.f, 0.f};
    int count = 0;
    for (int t = 0; t < NTOK; ++t) {
        if (s_match[t]) {
            float4 v = *(const float4*)(token_row(img, b, start, t) + col);
            acc.x += v.x; acc.y += v.y; acc.z += v.z; acc.w += v.w;
            ++count;
        }
    }

    float4 res;
    if (count > 0) {
        float fc = (float)count;     // sums / counts, like the reference
        res.x = acc.x / fc; res.y = acc.y / fc; res.z = acc.z / fc; res.w = acc.w / fc;
    } else {
        const float* ctr = token_row(img, b, start, topk[c]);
        res = *(const float4*)(ctr + col);
    }
    *(float4*)(out + (size_t)(b * OUTT + T_ + g * SG + c) * D_ + col) = res;
}

// ---------------------------------------------------------------------------
extern "C" void kernel_launch(void* const* d_in, const int* in_sizes, int n_in,
                              void* d_out, int out_size, void* d_ws, size_t ws_size,
                              hipStream_t stream) {
    const float* img = (const float*)d_in[0];
    float* out = (float*)d_out;

    // workspace carve-up (~296 KB total)
    float* ws_norm     = (float*)d_ws;                 // 32*768 f32
    int*   ws_valid    = (int*)(ws_norm + 32 * NTOK);  // 32*768 i32
    int*   ws_assigned = ws_valid + 32 * NTOK;         // 32*768 i32
    int*   ws_topk     = ws_assigned + 32 * NTOK;      // 32*64  i32

    copy_cls_kernel<<<(B_ * T_ * (D_ / 4) + 255) / 256, 256, 0, stream>>>(img, out);
    prep_kernel<<<32, 256, 0, stream>>>(img, ws_norm, ws_valid, ws_topk);
    simgemm_kernel<<<dim3(48, NG, B_), 32, 0, stream>>>(img, ws_norm, ws_topk, ws_assigned);
    pool_kernel<<<dim3(SG, NG, B_), 256, 0, stream>>>(img, ws_valid, ws_topk, ws_assigned, out);
}